// O3TensorProduct_19937238188635
// MI455X (gfx1250) — compile-verified
//
#include <hip/hip_runtime.h>

// ---------------------------------------------------------------------------
// O3 tensor product (e3nn 0e/1o x 0e/1o) + equivariant linear, f32 WMMA core.
//
// Math (per row n):
//   out_0e      = y0e * (x0e @ w_ss) + (seg_vv @ w_vv) + b
//   out_1o[:,i] = y1o_i * (x0e @ w_sv) + y0e * (x1o[:,:,i] @ w_vs)
//   seg_vv[u]   = INV_SQRT3 * dot(x1o[u,:], y1o)
//
// 6 GEMMs of (16 x 128) @ (128 x 128) per 16-row tile via
// V_WMMA_F32_16X16X4_F32 (full f32 precision; f32 matrix ceiling ~matches the
// 23.3 TB/s HBM roofline at AI~64, so no precision downgrade needed).
// Weights live in LDS (staged once per persistent block); per-tile A panels
// are staged with GLOBAL_LOAD_ASYNC_TO_LDS (ASYNCcnt) and the next tile's
// panel is prefetched (global_prefetch_b8) during the WMMA phase.
// ---------------------------------------------------------------------------

typedef float v2f __attribute__((ext_vector_type(2)));
typedef float v8f __attribute__((ext_vector_type(8)));
typedef int   v4i __attribute__((ext_vector_type(4)));

#define N_ROWS     100000
#define MULC       128
#define TILE_M     16
#define LDW        132            // padded LDS row stride (floats)
#define INV_SQRT3F 0.57735026918962584f

// LDS layout (float offsets)
#define OFF_WSS   0
#define OFF_WVV   (OFF_WSS + 128*LDW)
#define OFF_WSV   (OFF_WVV + 128*LDW)
#define OFF_WVS   (OFF_WSV + 128*LDW)
#define OFF_BIAS  (OFF_WVS + 128*LDW)
#define OFF_X0    (OFF_BIAS + 128)
#define OFF_VV    (OFF_X0 + TILE_M*LDW)
#define OFF_X1    (OFF_VV + TILE_M*LDW)          // 3 planes of TILE_M*LDW
#define OFF_Y0    (OFF_X1 + 3*TILE_M*LDW)
#define OFF_Y1    (OFF_Y0 + TILE_M)
#define SMEM_FLOATS (OFF_Y1 + TILE_M*3)
#define SMEM_BYTES  (SMEM_FLOATS * 4)            // 313,344 B <= 320 KB WGP LDS

// ---- CDNA5 async global->LDS staging (ASYNCcnt path), guarded ----
#if defined(__gfx1250__) && \
    __has_builtin(__builtin_amdgcn_global_load_async_to_lds_b128) && \
    __has_builtin(__builtin_amdgcn_global_load_async_to_lds_b32)  && \
    __has_builtin(__builtin_amdgcn_s_wait_asynccnt)
#define USE_ASYNC_LDS 1
#else
#define USE_ASYNC_LDS 0
#endif

#if USE_ASYNC_LDS
typedef __attribute__((address_space(1))) int g_i32;
typedef __attribute__((address_space(3))) int l_i32;
typedef __attribute__((address_space(1))) v4i g_v4i;
typedef __attribute__((address_space(3))) v4i l_v4i;
#endif

__device__ __forceinline__ void g2l_b128(const float* g, float* l) {
#if USE_ASYNC_LDS
    __builtin_amdgcn_global_load_async_to_lds_b128((g_v4i*)g, (l_v4i*)l, 0, 0);
#else
    *(float4*)l = *(const float4*)g;
#endif
}

__device__ __forceinline__ void g2l_b32(const float* g, float* l) {
#if USE_ASYNC_LDS
    __builtin_amdgcn_global_load_async_to_lds_b32((g_i32*)g, (l_i32*)l, 0, 0);
#else
    *l = *g;
#endif
}

__device__ __forceinline__ void async_stage_wait() {
#if USE_ASYNC_LDS
    __builtin_amdgcn_s_wait_asynccnt(0);
#endif
}

__global__ __launch_bounds__(256, 1)
void o3_tp_wmma_kernel(const float* __restrict__ x0e,
                       const float* __restrict__ x1o,
                       const float* __restrict__ y0e,
                       const float* __restrict__ y1o,
                       const float* __restrict__ wss,
                       const float* __restrict__ wvv,
                       const float* __restrict__ wsv,
                       const float* __restrict__ wvs,
                       const float* __restrict__ bias,
                       float* __restrict__ out)
{
    extern __shared__ float lds[];
    const int tid  = threadIdx.x;
    const int lane = tid & 31;
    const int wave = tid >> 5;

    // ---- stage weights + bias into LDS once per (persistent) block ----
    {
        const float* src[4] = {wss, wvv, wsv, wvs};
        float*       dst[4] = {lds + OFF_WSS, lds + OFF_WVV,
                               lds + OFF_WSV, lds + OFF_WVS};
        for (int m4 = 0; m4 < 4; ++m4) {
            for (int i = tid; i < 128 * 32; i += 256) {   // 4096 float4 / matrix
                int k = i >> 5;            // row (input channel u)
                int n = (i & 31) << 2;     // col (output channel w)
                g2l_b128(src[m4] + (size_t)i * 4, dst[m4] + k * LDW + n);
            }
        }
        for (int i = tid; i < 128; i += 256)
            g2l_b32(bias + i, lds + OFF_BIAS + i);
    }
    async_stage_wait();
    __syncthreads();

    // ---- per-wave constants for the WMMA fragment layout ----
    const int nt   = wave;                 // 16-column output slice
    const int n2   = lane & 15;
    const int ncol = nt * 16 + n2;         // output channel this lane covers
    const int kb0  = (lane >> 4) << 1;     // K sub-offset {0,2} per half-wave
    const int mA   = lane & 15;            // A-matrix row this lane covers
    const float bn = lds[OFF_BIAS + ncol];

    const int nTiles = N_ROWS / TILE_M;    // 6250 exact
    for (int rt = blockIdx.x; rt < nTiles; rt += gridDim.x) {
        const int r0 = rt * TILE_M;

        // ---- stage A operands for this 16-row tile (async -> LDS) ----
        {   // x_0e tile: 16 x 128 contiguous
            const float* s = x0e + (size_t)r0 * MULC;
            for (int i = tid; i < TILE_M * 32; i += 256) {
                int m = i >> 5, u = (i & 31) << 2;
                g2l_b128(s + (size_t)i * 4, lds + OFF_X0 + m * LDW + u);
            }
        }
        {   // x_1o tile: 16 x 128 x 3 contiguous -> de-interleave into 3 planes
            const float* s = x1o + (size_t)r0 * MULC * 3;
            for (int i = tid; i < TILE_M * 96; i += 256) {  // 6144 elements / 4
                int e   = i << 2;
                int m   = e / 384;
                int rem = e - m * 384;
                #pragma unroll
                for (int j = 0; j < 4; ++j) {
                    int u = (rem + j) / 3;
                    int c = (rem + j) - u * 3;
                    g2l_b32(s + e + j,
                            lds + OFF_X1 + c * (TILE_M * LDW) + m * LDW + u);
                }
            }
        }
        if (tid < TILE_M) {  // y scalars (tiny, plain DS stores)
            lds[OFF_Y0 + tid] = y0e[r0 + tid];
            lds[OFF_Y1 + tid * 3 + 0] = y1o[(size_t)(r0 + tid) * 3 + 0];
            lds[OFF_Y1 + tid * 3 + 1] = y1o[(size_t)(r0 + tid) * 3 + 1];
            lds[OFF_Y1 + tid * 3 + 2] = y1o[(size_t)(r0 + tid) * 3 + 2];
        }
        async_stage_wait();
        __syncthreads();

        // seg_vv[m][u] = INV_SQRT3 * dot(x1o[m,u,:], y1o[m,:])
        for (int i = tid; i < TILE_M * MULC; i += 256) {
            int m = i >> 7, u = i & 127;
            float s = lds[OFF_X1 + 0 * (TILE_M * LDW) + m * LDW + u] * lds[OFF_Y1 + m * 3 + 0]
                    + lds[OFF_X1 + 1 * (TILE_M * LDW) + m * LDW + u] * lds[OFF_Y1 + m * 3 + 1]
                    + lds[OFF_X1 + 2 * (TILE_M * LDW) + m * LDW + u] * lds[OFF_Y1 + m * 3 + 2];
            lds[OFF_VV + m * LDW + u] = s * INV_SQRT3F;
        }
        __syncthreads();

        // Prefetch next tile's A panel into cache while WMMAs run.
        {
            const int rtn = rt + gridDim.x;
            if (rtn < nTiles) {
                const char* p0 = (const char*)(x0e + (size_t)rtn * MULC);
                const char* p1 = (const char*)(x1o + (size_t)rtn * MULC * 3);
                if (tid < 64) __builtin_prefetch(p0 + (size_t)tid * 128);  // 8 KB
                __builtin_prefetch(p1 + (size_t)tid * 96);                 // 24 KB
            }
        }

        // ---- 6 accumulators, K = 128 in steps of 4 (V_WMMA_F32_16X16X4_F32)
        v8f acc_ss = {}, acc_vv = {}, acc_sv = {};
        v8f acc_q0 = {}, acc_q1 = {}, acc_q2 = {};

        const float* pA0  = lds + OFF_X0 + mA * LDW + kb0;
        const float* pAv  = lds + OFF_VV + mA * LDW + kb0;
        const float* pA1a = lds + OFF_X1 + 0 * (TILE_M * LDW) + mA * LDW + kb0;
        const float* pA1b = lds + OFF_X1 + 1 * (TILE_M * LDW) + mA * LDW + kb0;
        const float* pA1c = lds + OFF_X1 + 2 * (TILE_M * LDW) + mA * LDW + kb0;
        const float* pBss = lds + OFF_WSS + kb0 * LDW + ncol;
        const float* pBvv = lds + OFF_WVV + kb0 * LDW + ncol;
        const float* pBsv = lds + OFF_WSV + kb0 * LDW + ncol;
        const float* pBvs = lds + OFF_WVS + kb0 * LDW + ncol;

        #pragma unroll 4
        for (int k = 0; k < MULC; k += 4) {
            v2f a0  = *(const v2f*)pA0;
            v2f av  = *(const v2f*)pAv;
            v2f a1a = *(const v2f*)pA1a;
            v2f a1b = *(const v2f*)pA1b;
            v2f a1c = *(const v2f*)pA1c;
            v2f bssf; bssf.x = pBss[0]; bssf.y = pBss[LDW];
            v2f bvvf; bvvf.x = pBvv[0]; bvvf.y = pBvv[LDW];
            v2f bsvf; bsvf.x = pBsv[0]; bsvf.y = pBsv[LDW];
            v2f bvsf; bvsf.x = pBvs[0]; bvsf.y = pBvs[LDW];

            acc_ss = __builtin_amdgcn_wmma_f32_16x16x4_f32(false, a0,  false, bssf, (short)0, acc_ss, false, false);
            acc_sv = __builtin_amdgcn_wmma_f32_16x16x4_f32(false, a0,  false, bsvf, (short)0, acc_sv, false, false);
            acc_vv = __builtin_amdgcn_wmma_f32_16x16x4_f32(false, av,  false, bvvf, (short)0, acc_vv, false, false);
            acc_q0 = __builtin_amdgcn_wmma_f32_16x16x4_f32(false, a1a, false, bvsf, (short)0, acc_q0, false, false);
            acc_q1 = __builtin_amdgcn_wmma_f32_16x16x4_f32(false, a1b, false, bvsf, (short)0, acc_q1, false, false);
            acc_q2 = __builtin_amdgcn_wmma_f32_16x16x4_f32(false, a1c, false, bvsf, (short)0, acc_q2, false, false);

            pA0 += 4; pAv += 4; pA1a += 4; pA1b += 4; pA1c += 4;
            pBss += 4 * LDW; pBvv += 4 * LDW; pBsv += 4 * LDW; pBvs += 4 * LDW;
        }

        // ---- epilogue: scale by y, add bias, store [0e | 1o] flat layout ----
        #pragma unroll
        for (int v = 0; v < 8; ++v) {
            const int m   = (lane < 16) ? v : (v + 8);   // C/D VGPR->row mapping
            const int row = r0 + m;
            const float y0  = lds[OFF_Y0 + m];
            const float y10 = lds[OFF_Y1 + m * 3 + 0];
            const float y11 = lds[OFF_Y1 + m * 3 + 1];
            const float y12 = lds[OFF_Y1 + m * 3 + 2];

            const size_t ob  = (size_t)row * 512;
            out[ob + ncol] = y0 * acc_ss[v] + acc_vv[v] + bn;

            const size_t ob1 = ob + 128 + (size_t)ncol * 3;
            const float psv = acc_sv[v];
            out[ob1 + 0] = y10 * psv + y0 * acc_q0[v];
            out[ob1 + 1] = y11 * psv + y0 * acc_q1[v];
            out[ob1 + 2] = y12 * psv + y0 * acc_q2[v];
        }
        __syncthreads();   // protect LDS A-staging before next tile
    }
}

extern "C" void kernel_launch(void* const* d_in, const int* in_sizes, int n_in,
                              void* d_out, int out_size, void* d_ws, size_t ws_size,
                              hipStream_t stream) {
    const float* x0e  = (const float*)d_in[0];
    const float* x1o  = (const float*)d_in[1];
    const float* y0e  = (const float*)d_in[2];
    const float* y1o  = (const float*)d_in[3];
    const float* wss  = (const float*)d_in[4];
    const float* wvv  = (const float*)d_in[5];
    const float* wsv  = (const float*)d_in[6];
    const float* wvs  = (const float*)d_in[7];
    const float* bias = (const float*)d_in[8];
    float* out = (float*)d_out;

    (void)in_sizes; (void)n_in; (void)out_size; (void)d_ws; (void)ws_size;

    // Allow >64KB dynamic LDS (gfx1250 WGP has 320KB).
    (void)hipFuncSetAttribute((const void*)o3_tp_wmma_kernel,
                              hipFuncAttributeMaxDynamicSharedMemorySize,
                              SMEM_BYTES);

    // Persistent blocks: weights staged once per block, grid-stride over tiles.
    o3_tp_wmma_kernel<<<512, 256, SMEM_BYTES, stream>>>(
        x0e, x1o, y0e, y1o, wss, wvv, wsv, wvs, bias, out);
}